// SeqMasking_2035814499079
// MI455X (gfx1250) — compile-verified
//
#include <hip/hip_runtime.h>
#include <hip/hip_bf16.h>
#include <stdint.h>

// SeqMasking: tail-compaction of kept rows, zero-filled head.
// Pure bandwidth problem: ~256MB moved -> ~11us at 23.3 TB/s.
// CDNA5 features: async global->LDS staging (ASYNCcnt + s_wait_asynccnt),
// wave32 block scan, branch-free output-partitioned copy with b128
// non-temporal loads/stores, unrolled x2 for deeper MLP.

typedef float v4f __attribute__((ext_vector_type(4)));

constexpr int   T_DIM   = 2048;
constexpr int   D_DIM   = 256;           // floats per row (1KB)
constexpr int   F4_ROW  = D_DIM / 4;     // 64 float4 per row
constexpr int   THREADS = 256;           // 8 waves (wave32)
constexpr int   NWAVES  = THREADS / 32;
constexpr int   SLICES  = 16;            // blocks per batch row
constexpr int   RPS     = T_DIM / SLICES;// output rows per slice = 128
constexpr float P_DROP  = 0.15f;

__global__ __launch_bounds__(THREADS)
void seqmask_compact_kernel(const float* __restrict__ x,
                            const float* __restrict__ rnd,
                            float* __restrict__ out)
{
    __shared__ float s_rand[T_DIM];   // 8KB, staged via async DMA
    __shared__ int   s_src[T_DIM];    // 8KB: rank r -> source row t
    __shared__ int   s_sum[THREADS];  // 1KB scan scratch

    const int b     = blockIdx.y;
    const int slice = blockIdx.x;
    const int tid   = threadIdx.x;
    const int lane  = tid & 31;
    const int wid   = tid >> 5;

    // ---- Phase 1: async-stage this batch's rand row into LDS (ASYNCcnt path) ----
    {
        const void* g    = (const void*)(rnd + (size_t)b * T_DIM + tid * 8);
        uint32_t    loff = (uint32_t)(size_t)(void*)&s_rand[tid * 8];
        // INST_OFFSET applies to both LDS and global address -> one base covers
        // both 16B halves of this thread's 8 floats.
        asm volatile(
            "global_load_async_to_lds_b128 %0, %1, off\n\t"
            "global_load_async_to_lds_b128 %0, %1, off offset:16\n\t"
            "s_wait_asynccnt 0"
            :: "v"(loff), "v"(g) : "memory");
    }
    __syncthreads();   // every wave drained its own ASYNCcnt -> LDS fully populated

    // ---- Phase 2: keep mask + block-wide inclusive scan (8 elems/thread) ----
    int kbits = 0, lsum = 0;
    #pragma unroll
    for (int j = 0; j < 8; ++j) {
        int k = (s_rand[tid * 8 + j] > P_DROP) ? 1 : 0;
        kbits |= k << j;
        lsum  += k;
    }
    s_sum[tid] = lsum;
    #pragma unroll
    for (int off = 1; off < THREADS; off <<= 1) {
        __syncthreads();
        int add = (tid >= off) ? s_sum[tid - off] : 0;
        __syncthreads();
        s_sum[tid] += add;
    }
    __syncthreads();
    const int total = s_sum[THREADS - 1];      // l = kept tokens in this row
    const int excl  = s_sum[tid] - lsum;       // exclusive prefix for this thread

    // rank -> source-row table (each rank written exactly once block-wide)
    int cs = excl;
    #pragma unroll
    for (int j = 0; j < 8; ++j) {
        if ((kbits >> j) & 1) {
            s_src[cs] = tid * 8 + j;           // cs is 0-based rank here
            ++cs;
        }
    }
    __syncthreads();

    // ---- Phase 3: output-partitioned writes for this slice ----
    // Output row o:  o <  zsplit -> zeros
    //                o >= zsplit -> copy of x[:, s_src[o - zsplit], :]
    const int    zsplit = T_DIM - total;
    const size_t base4  = ((size_t)b * T_DIM * D_DIM) >> 2;  // batch base, float4 units
    const v4f* __restrict__ x4 = (const v4f*)x;
    v4f* __restrict__ o4       = (v4f*)out;
    const int t0 = slice * RPS;

    // zero head rows in this slice: pure stores, no stalls
    {
        const int zhi = min(t0 + RPS, zsplit);
        const v4f zero = (v4f){0.f, 0.f, 0.f, 0.f};
        for (int o = t0 + wid; o < zhi; o += NWAVES) {
            v4f* dp = o4 + base4 + (size_t)o * F4_ROW;
            __builtin_nontemporal_store(zero, dp + lane);
            __builtin_nontemporal_store(zero, dp + lane + 32);
        }
    }

    // copy tail rows in this slice: branch-free body, unrolled for MLP
    {
        const int clo = max(t0, zsplit);
        const int chi = t0 + RPS;
        #pragma unroll 2
        for (int o = clo + wid; o < chi; o += NWAVES) {
            const int src = s_src[o - zsplit];   // wave-uniform LDS broadcast
            const v4f* sp = x4 + base4 + (size_t)src * F4_ROW;
            v4f*       dp = o4 + base4 + (size_t)o   * F4_ROW;
            v4f a = __builtin_nontemporal_load(sp + lane);
            v4f c = __builtin_nontemporal_load(sp + lane + 32);
            __builtin_nontemporal_store(a, dp + lane);
            __builtin_nontemporal_store(c, dp + lane + 32);
        }
    }
}

extern "C" void kernel_launch(void* const* d_in, const int* in_sizes, int n_in,
                              void* d_out, int out_size, void* d_ws, size_t ws_size,
                              hipStream_t stream) {
    const float* x   = (const float*)d_in[0];   // (B, T, D) f32
    const float* rnd = (const float*)d_in[1];   // (B, T)    f32
    float*       out = (float*)d_out;           // (B, T, D) f32

    const int B = in_sizes[1] / T_DIM;          // 64 for the reference shapes

    dim3 grid(SLICES, B);
    dim3 block(THREADS);
    seqmask_compact_kernel<<<grid, block, 0, stream>>>(x, rnd, out);
}